// VariationalBayesRouting2d_1726576853904
// MI455X (gfx1250) — compile-verified
//
#include <hip/hip_runtime.h>
#include <math.h>
#include <stdint.h>

typedef float v2f __attribute__((ext_vector_type(2)));
typedef float v8f __attribute__((ext_vector_type(8)));

#define NB 2          // batch
#define BCAPS 32      // input capsules
#define CCAPS 32      // output capsules
#define DD 16         // pose dim D = P*P
#define FF 12         // spatial F
#define KK 9          // K*K
#define FS 144        // F*F
#define HH 14         // (F-1)*S + K
#define ROW 108       // F*K*K contiguous per f1 (f2*9+k1*3+k2)
#define SLAB 1728     // DD * ROW
#define PER_BIC 1296  // FF*ROW per (b,i,c)

// V_ji strides in elements, layout (b, B, C, D, 1, F, F, K, K)
#define VS_B  21233664
#define VS_I  663552
#define VS_C  20736
#define VS_D  1296
#define VS_F1 108

// ---- CDNA5 async global->LDS copy (GLOBAL_LOAD_ASYNC_TO_LDS_B128, GV mode) ----
// LDS byte address = low 32 bits of the generic shared pointer (flat->LDS mapping
// truncates to addr[31:0]); tracked with ASYNCcnt.
__device__ __forceinline__ void async_ld16(void* lds_dst, const void* gsrc) {
  unsigned lo = (unsigned)(uintptr_t)lds_dst;
  unsigned long long ga = (unsigned long long)(uintptr_t)gsrc;
  asm volatile("global_load_async_to_lds_b128 %0, %1, off"
               :: "v"(lo), "v"(ga) : "memory");
}
__device__ __forceinline__ void async_wait0() {
  asm volatile("s_wait_asynccnt 0" ::: "memory");
}

__device__ __forceinline__ float digammaf_(float x) {
  // valid for x > 0; shift to x>=6 then asymptotic series
  float r = 0.f;
  #pragma unroll 1
  while (x < 6.f) { r -= 1.f / x; x += 1.f; }
  float inv = 1.f / x, inv2 = inv * inv;
  return r + logf(x) - 0.5f * inv
         - inv2 * (0.0833333333f - inv2 * (0.0083333333f - inv2 * 0.0039682540f));
}

__global__ void k_init_R(float* __restrict__ R, long n) {
  long i = (long)blockIdx.x * blockDim.x + threadIdx.x;
  if (i < n) R[i] = 1.0f / CCAPS;
}

// One block per (b,c,f1): accumulate R_j, S1[d]=Sum Rw*V, S2[d]=Sum Rw*V^2 over (i,kk),
// then site stats: m, 1/invPsi, Elnlambda, alpha, and the per-site lnp constant.
__global__ void __launch_bounds__(256)
k_stats(const float* __restrict__ V, const float* __restrict__ Ai,
        const float* __restrict__ R,
        float* __restrict__ STm, float* __restrict__ STw0, float* __restrict__ STs) {
  const int bx = blockIdx.x;
  const int f1 = bx % FF;
  const int c  = (bx / FF) % CCAPS;
  const int b  = bx / (FF * CCAPS);
  const int t  = threadIdx.x;

  __shared__ __align__(16) float Vs[SLAB];
  __shared__ float rw[ROW];
  __shared__ float rjs[FF];
  __shared__ float redA[192], redB[192], redC[192];

  const int f2 = t >> 4;   // for t<192
  const int d  = t & 15;

  float s1 = 0.f, s2 = 0.f, rj = 0.f;
  const int vbase = b * VS_B + c * VS_C + f1 * VS_F1;

  for (int i = 0; i < BCAPS; ++i) {
    const float* vsl = V + vbase + i * VS_I;
    for (int u = t; u < 432; u += 256) {              // 432 x b128 = 1728 floats
      int dd = u / 27, q = u % 27;
      async_ld16(&Vs[(dd * 27 + q) * 4], vsl + dd * VS_D + q * 4);
      if (i + 1 < BCAPS)
        __builtin_prefetch(vsl + VS_I + dd * VS_D + q * 4, 0, 3);
    }
    if (t < ROW) {
      int rbase = ((b * BCAPS + i) * CCAPS + c) * PER_BIC + f1 * ROW;
      int abase = (b * BCAPS + i) * PER_BIC + f1 * ROW;
      rw[t] = R[rbase + t] * Ai[abase + t];
    }
    async_wait0();
    __syncthreads();
    if (t < 192) {
      const float* vr = &Vs[d * ROW + f2 * KK];
      const float* rr = &rw[f2 * KK];
      #pragma unroll
      for (int k = 0; k < KK; ++k) {
        float rv = rr[k], vv = vr[k];
        s1 = fmaf(rv, vv, s1);
        s2 = fmaf(rv * vv, vv, s2);
        if (d == 0) rj += rv;
      }
    }
    __syncthreads();
  }
  if (t < 192 && d == 0) rjs[f2] = rj;
  __syncthreads();

  const int sidx0 = (b * CCAPS + c) * FS + f1 * FF;
  if (t < 192) {
    float Rj  = rjs[f2];
    float kap = 1.f + Rj;          // KAPPA0 + R_j
    float nu  = 17.f + Rj;         // NU0 + R_j
    float mu  = s1 / Rj;
    float m   = s1 / kap;          // m_j = R_j*mu/kappa = S1/kappa
    float sig = s2 - s1 * mu;      // S2 - R_j*mu^2
    float invPsi = 1.f + sig + (Rj / kap) * mu * mu;  // PSI0_DIAG = 1
    float w0 = 1.f / invPsi;
    STm [(sidx0 + f2) * DD + d] = m;
    STw0[(sidx0 + f2) * DD + d] = w0;
    redA[t] = logf(invPsi);
    redB[t] = digammaf_(0.5f * (nu - (float)d));
    redC[t] = m * m * w0;
  }
  __syncthreads();
  if (t < FF) {
    float Rj  = rjs[t];
    float kap = 1.f + Rj;
    float nu  = 17.f + Rj;
    float alpha = 1.f + Rj;        // ALPHA0 + R_j
    float slog = 0.f, sdg = 0.f, ct = 0.f;
    #pragma unroll
    for (int dd = 0; dd < DD; ++dd) {
      slog += redA[t * DD + dd];
      sdg  += redB[t * DD + dd];
      ct   += redC[t * DD + dd];
    }
    float lndet  = -slog;
    float Elnlam = sdg + 16.f * 0.69314718056f + lndet;   // + D*ln2 + lndet
    float halfnu = 0.5f * nu;
    // Csite = dg(alpha) + 0.5*Elnlam - 0.5*D*ln(2pi) - 0.5*D/kappa - halfnu*Sum(m^2/invPsi)
    float Cs = digammaf_(alpha) + 0.5f * Elnlam - 8.f * 1.83787706641f
               - 8.f / kap - halfnu * ct;
    int sidx = sidx0 + t;
    STs[sidx * 8 + 0] = Cs;
    STs[sidx * 8 + 1] = halfnu;
    STs[sidx * 8 + 2] = alpha;
    STs[sidx * 8 + 3] = Elnlam;
  }
}

// digamma of per-(b,f1,f2) alpha sum over C
__global__ void k_dgsum(const float* __restrict__ STs, float* __restrict__ DG) {
  int t = threadIdx.x;
  if (t >= NB * FS) return;
  int b = t / FS, s = t % FS;
  float sum = 0.f;
  for (int c = 0; c < CCAPS; ++c)
    sum += STs[((b * CCAPS + c) * FS + s) * 8 + 2];
  DG[t] = digammaf_(sum);
}

// ElnQ pass with V_WMMA_F32_16X16X4_F32: per (f2) site,
//  A (16x4) rows = kernel positions kk (9 valid, rows 9-15 are don't-care),
//  B col0 = m/invPsi, cols 1..15 = 1/invPsi (only cols 0,1 of D are read).
//  acc1 = Sum_d V*B (col0 used), acc2 = Sum_d V^2*B (col1 used).
__global__ void __launch_bounds__(256)
k_p(const float* __restrict__ V,
    const float* __restrict__ STm, const float* __restrict__ STw0,
    const float* __restrict__ STs, const float* __restrict__ DG,
    float* __restrict__ P, float* __restrict__ FULLB) {
  const int bx = blockIdx.x;
  const int f1 = bx % FF;
  const int c  = (bx / FF) % CCAPS;
  const int b  = bx / (FF * CCAPS);
  const int t  = threadIdx.x;
  const int lane  = t & 31;
  const int wid   = t >> 5;
  const int m16   = lane & 15;   // A row / B column index
  const int khalf = lane >> 4;   // 0: K=0,1 ; 1: K=2,3 (per 16x16x4 f32 layout)

  __shared__ __align__(16) float Vs[SLAB];
  __shared__ float WL[FF * 32];  // [f2][0..15]=w1=m/invPsi, [f2][16..31]=w0=1/invPsi
  __shared__ float cs[FF], hn[FF];
  __shared__ float qA[ROW], qB[ROW];

  const int sidx0 = (b * CCAPS + c) * FS + f1 * FF;
  if (t < 192) {
    int f2 = t >> 4, d = t & 15;
    float w0 = STw0[(sidx0 + f2) * DD + d];
    WL[f2 * 32 + 16 + d] = w0;
    WL[f2 * 32 + d] = STm[(sidx0 + f2) * DD + d] * w0;
  }
  if (t < FF) {
    cs[t] = STs[(sidx0 + t) * 8 + 0] - DG[b * FS + f1 * FF + t];
    hn[t] = STs[(sidx0 + t) * 8 + 1];
  }
  __syncthreads();

  const int vbase = b * VS_B + c * VS_C + f1 * VS_F1;
  const int arow = (m16 < 9) ? m16 : 8;       // clamp: rows 9-15 are don't-care
  const int bsel = (m16 == 0) ? 0 : 16;       // col0 -> w1, cols>=1 -> w0

  for (int i = 0; i < BCAPS; ++i) {
    const float* vsl = V + vbase + i * VS_I;
    for (int u = t; u < 432; u += 256) {
      int dd = u / 27, q = u % 27;
      async_ld16(&Vs[(dd * 27 + q) * 4], vsl + dd * VS_D + q * 4);
    }
    async_wait0();
    __syncthreads();

    for (int f2 = wid; f2 < FF; f2 += 8) {     // wave-uniform: EXEC all-ones at WMMA
      v8f acc1 = {0.f,0.f,0.f,0.f,0.f,0.f,0.f,0.f};
      v8f acc2 = {0.f,0.f,0.f,0.f,0.f,0.f,0.f,0.f};
      const int aoff = f2 * KK + arow;
      const int boff = f2 * 32 + bsel;
      #pragma unroll
      for (int ks = 0; ks < 4; ++ks) {
        int d0 = ks * 4 + 2 * khalf;
        float a0 = Vs[d0 * ROW + aoff];
        float a1 = Vs[(d0 + 1) * ROW + aoff];
        float b0 = WL[boff + d0];
        float b1 = WL[boff + d0 + 1];
        v2f av  = {a0, a1};
        v2f av2 = {a0 * a0, a1 * a1};
        v2f bv  = {b0, b1};
        acc1 = __builtin_amdgcn_wmma_f32_16x16x4_f32(false, av,  false, bv, (short)0, acc1, false, false);
        acc2 = __builtin_amdgcn_wmma_f32_16x16x4_f32(false, av2, false, bv, (short)0, acc2, false, false);
      }
      // D layout: VGPR j, lanes 0-15 -> (M=j, N=lane); lanes 16-31 -> (M=j+8, N=lane-16)
      if (lane == 0) {
        #pragma unroll
        for (int j = 0; j < 8; ++j) qA[f2 * KK + j] = acc1[j];
      }
      if (lane == 16) qA[f2 * KK + 8] = acc1[0];
      if (lane == 1) {
        #pragma unroll
        for (int j = 0; j < 8; ++j) qB[f2 * KK + j] = acc2[j];
      }
      if (lane == 17) qB[f2 * KK + 8] = acc2[0];
    }
    __syncthreads();

    if (t < ROW) {
      int f2 = t / KK, kk = t % KK;
      float q = qB[t] - 2.f * qA[t];           // + cterm folded into cs
      float p = expf(cs[f2] - hn[f2] * q);
      int pbase = ((b * BCAPS + i) * CCAPS + c) * PER_BIC + f1 * ROW;
      P[pbase + t] = p;
      int k1 = kk / 3, k2 = kk % 3;
      atomicAdd(&FULLB[((b * BCAPS + i) * HH + (f1 + k1)) * HH + (f2 + k2)], p);
    }
    __syncthreads();
  }
}

__global__ void k_norm(const float* __restrict__ P, const float* __restrict__ FULLB,
                       float* __restrict__ R) {
  long idx = (long)blockIdx.x * blockDim.x + threadIdx.x;
  const long n = (long)NB * BCAPS * CCAPS * PER_BIC;
  if (idx >= n) return;
  int r  = (int)(idx % PER_BIC);
  long bic = idx / PER_BIC;
  long bi  = bic / CCAPS;
  int f1 = r / ROW;
  int r2 = r % ROW;
  int f2 = r2 / KK, kk = r2 % KK;
  int k1 = kk / 3, k2 = kk % 3;
  float sump = FULLB[(bi * HH + f1 + k1) * HH + (f2 + k2)];
  R[idx] = P[idx] / fmaxf(sump, 1e-11f);
}

// One block per output capsule c: activation + per-capsule batch-norm + sigmoid, pose batch-norm.
__global__ void __launch_bounds__(256)
k_final(const float* __restrict__ STm, const float* __restrict__ STs,
        const float* __restrict__ DG,
        const float* __restrict__ bu, const float* __restrict__ ba,
        float* __restrict__ out) {
  const int c = blockIdx.x;
  const int t = threadIdx.x;
  __shared__ float As[NB * FS];
  __shared__ float r0[256], r1[256], r2[256], r3[256];

  const float Hqc = 8.f * (1.83787706641f + 1.f);   // 0.5*D*ln(2*pi*e)
  float pa = 0.f, pa2 = 0.f, pm = 0.f, pm2 = 0.f;

  for (int s = t; s < NB * FS; s += 256) {
    int b = s / FS, s2 = s % FS;
    int sidx = (b * CCAPS + c) * FS + s2;
    float alpha = STs[sidx * 8 + 2];
    float Elnl  = STs[sidx * 8 + 3];
    float Elnpi = digammaf_(alpha) - DG[s];
    float Hq = Hqc - 0.5f * Elnl;
    float a = ba[c] - (expf(Elnpi) * Hq + bu[c]);
    As[s] = a;
    pa += a; pa2 += a * a;
  }
  for (int u = t; u < NB * FS * DD; u += 256) {
    int b = u / (FS * DD);
    int rem = u % (FS * DD);
    int d = rem / FS, s2 = rem % FS;
    float m = STm[((b * CCAPS + c) * FS + s2) * DD + d];
    pm += m; pm2 += m * m;
  }
  r0[t] = pa; r1[t] = pa2; r2[t] = pm; r3[t] = pm2;
  __syncthreads();
  for (int off = 128; off > 0; off >>= 1) {
    if (t < off) {
      r0[t] += r0[t + off]; r1[t] += r1[t + off];
      r2[t] += r2[t + off]; r3[t] += r3[t + off];
    }
    __syncthreads();
  }
  const float na = (float)(NB * FS), nm = (float)(NB * FS * DD);
  float ma = r0[0] / na, va = r1[0] / na - ma * ma;
  float mv = r2[0] / nm, vv = r3[0] / nm - mv * mv;
  float ia = rsqrtf(va + 1e-5f), im = rsqrtf(vv + 1e-5f);

  for (int s = t; s < NB * FS; s += 256) {
    int b = s / FS, s2 = s % FS;
    float x = (As[s] - ma) * ia;
    out[(b * CCAPS + c) * FS + s2] = 1.f / (1.f + expf(-x));
  }
  for (int u = t; u < NB * FS * DD; u += 256) {
    int b = u / (FS * DD);
    int rem = u % (FS * DD);
    int d = rem / FS, s2 = rem % FS;
    float m = STm[((b * CCAPS + c) * FS + s2) * DD + d];
    out[NB * CCAPS * FS + ((b * CCAPS + c) * DD + d) * FS + s2] = (m - mv) * im;
  }
}

extern "C" void kernel_launch(void* const* d_in, const int* in_sizes, int n_in,
                              void* d_out, int out_size, void* d_ws, size_t ws_size,
                              hipStream_t stream) {
  (void)in_sizes; (void)n_in; (void)out_size; (void)ws_size;
  const float* Ai = (const float*)d_in[0];
  const float* V  = (const float*)d_in[1];
  const float* bu = (const float*)d_in[2];
  const float* ba = (const float*)d_in[3];
  float* out = (float*)d_out;
  float* W   = (float*)d_ws;

  const long NR = (long)NB * BCAPS * CCAPS * PER_BIC;  // 2,654,208
  float* Rb    = W;
  float* Pb    = Rb + NR;
  float* STm   = Pb + NR;                 // 9216*16
  float* STw   = STm + (long)NB * CCAPS * FS * DD;
  float* STs   = STw + (long)NB * CCAPS * FS * DD;    // 9216*8
  float* DG    = STs + (long)NB * CCAPS * FS * 8;     // 288
  float* FULLB = DG + NB * FS;                        // 12544

  k_init_R<<<(int)((NR + 255) / 256), 256, 0, stream>>>(Rb, NR);
  const int NBLK = NB * CCAPS * FF;   // 768
  for (int it = 0; it < 3; ++it) {
    k_stats<<<NBLK, 256, 0, stream>>>(V, Ai, Rb, STm, STw, STs);
    k_dgsum<<<1, NB * FS, 0, stream>>>(STs, DG);
    if (it < 2) {
      hipMemsetAsync(FULLB, 0, (size_t)NB * BCAPS * HH * HH * sizeof(float), stream);
      k_p<<<NBLK, 256, 0, stream>>>(V, STm, STw, STs, DG, Pb, FULLB);
      k_norm<<<(int)((NR + 255) / 256), 256, 0, stream>>>(Pb, FULLB, Rb);
    }
  }
  k_final<<<CCAPS, 256, 0, stream>>>(STm, STs, DG, bu, ba, out);
}